// NSS_12249246728744
// MI455X (gfx1250) — compile-verified
//
#include <hip/hip_runtime.h>
#include <math.h>

typedef __attribute__((ext_vector_type(2))) float v2f;
typedef __attribute__((ext_vector_type(8))) float v8f;

#define BATCH 4096
#define IN    128
#define N1    256
#define N2    64
#define ITERS 100
#define THR   0.1f
#define GAMMA 0.1f
#define INV_SCALE 1.25f
#define LR_OVER_B (0.1f / 4096.0f)

__device__ __forceinline__ v8f wmma4(v2f a, v2f b, v8f c) {
    return __builtin_amdgcn_wmma_f32_16x16x4_f32(false, a, false, b, (short)0, c,
                                                 false, false);
}

__device__ __forceinline__ float tdq(float u) {
    float au = fabsf(u) - THR;
    float soft = (au > 0.0f) ? copysignf(au, u) : 0.0f;
    float q = rintf(soft / THR);
    q = fminf(fmaxf(q, -3.0f), 3.0f) * THR;
    return q;
}

// ---------------- prologue kernels ----------------

__global__ void gram1_kernel(const float* __restrict__ d1, float* __restrict__ g1) {
    int idx = blockIdx.x * blockDim.x + threadIdx.x;       // 65536
    int i = idx >> 8, j = idx & 255;
    float acc = 0.0f;
    for (int k = 0; k < IN; ++k)
        acc += d1[k * N1 + i] * d1[k * N1 + j];
    g1[idx] = acc - ((i == j) ? 1.0f : 0.0f);
}

__global__ void gram2_kernel(const float* __restrict__ d2, float* __restrict__ g2) {
    int idx = blockIdx.x * blockDim.x + threadIdx.x;       // 4096
    int i = idx >> 6, j = idx & 63;
    float acc = 0.0f;
    for (int k = 0; k < N1; ++k)
        acc += d2[k * N2 + i] * d2[k * N2 + j];
    g2[idx] = acc - ((i == j) ? 1.0f : 0.0f);
}

__global__ void b1_kernel(const float* __restrict__ x, const float* __restrict__ d1,
                          float* __restrict__ b1) {
    int idx = blockIdx.x * blockDim.x + threadIdx.x;       // 4096*256
    int s = idx >> 8, j = idx & 255;
    float acc = 0.0f;
    for (int k = 0; k < IN; ++k)
        acc += x[s * IN + k] * d1[k * N1 + j];
    b1[idx] = acc;
}

__global__ void t1_kernel(const float* __restrict__ d1, float* __restrict__ d1t) {
    int idx = blockIdx.x * blockDim.x + threadIdx.x;       // 256*128, d1t[j][i]
    int i = idx & 127, j = idx >> 7;
    d1t[idx] = d1[i * N1 + j];
}

__global__ void t2_kernel(const float* __restrict__ d2, float* __restrict__ d2t) {
    int idx = blockIdx.x * blockDim.x + threadIdx.x;       // 64*256, d2t[j][i]
    int i = idx & 255, j = idx >> 8;
    d2t[idx] = d2[i * N2 + j];
}

// ---------------- main persistent LCA kernel ----------------

__global__ __launch_bounds__(256) void nss_main_kernel(
    const float* __restrict__ g1, const float* __restrict__ g2,
    const float* __restrict__ b1g, const float* __restrict__ d2g,
    float* __restrict__ dec1o, float* __restrict__ dec2o,
    float* __restrict__ ns1o, float* __restrict__ ns2o,
    float* __restrict__ a1w, float* __restrict__ a2w) {

    __shared__ float a1s[16 * 257];   // current a1 tile (padded)
    __shared__ float m1s[16 * 257];   // a1@G1 result / cnt scratch
    __shared__ float a2s[16 * 65];
    __shared__ float m2s[16 * 65];    // scaled@D2
    __shared__ float m2b[16 * 65];    // a2@G2

    const int tid  = threadIdx.x;
    const int wave = tid >> 5;
    const int lane = tid & 31;
    const int m_a  = lane & 15;              // A-frag / D-frag N index
    const int kb   = (lane >> 4) << 1;       // K sub-offset (0 or 2)
    const int n_b  = lane & 15;
    const int row0 = blockIdx.x << 4;        // 16 rows per block
    const int c2   = tid & 63;
    const int rq   = tid >> 6;

    float u1r[16], dec1r[16], cnt1r[16], b1r[16];
    float u2r[4], dec2r[4], cnt2r[4];

#pragma unroll
    for (int r = 0; r < 16; ++r) {
        u1r[r] = 0.0f; dec1r[r] = 0.0f; cnt1r[r] = 0.0f;
        b1r[r] = b1g[(row0 + r) * N1 + tid];
        a1s[r * 257 + tid] = 0.0f;
    }
#pragma unroll
    for (int j = 0; j < 4; ++j) { u2r[j] = 0.0f; dec2r[j] = 0.0f; cnt2r[j] = 0.0f; }
    for (int idx = tid; idx < 16 * 64; idx += 256)
        a2s[(idx >> 6) * 65 + (idx & 63)] = 0.0f;
    __syncthreads();

    for (int it = 0; it < ITERS; ++it) {
        // ---- layer 1: M1 = a1 @ G1 (16x256, G1 symmetric) ----
#pragma unroll
        for (int tp = 0; tp < 2; ++tp) {
            const int nt = wave + tp * 8;
            const int cb = nt * 16 + n_b;
            v8f c = (v8f)0.0f;
#pragma unroll 4
            for (int k0 = 0; k0 < N1; k0 += 4) {
                v2f a, b;
                a[0] = a1s[m_a * 257 + k0 + kb];
                a[1] = a1s[m_a * 257 + k0 + kb + 1];
                b[0] = g1[(k0 + kb) * N1 + cb];
                b[1] = g1[(k0 + kb + 1) * N1 + cb];
                c = wmma4(a, b, c);
            }
#pragma unroll
            for (int v = 0; v < 8; ++v)
                m1s[((lane >> 4) * 8 + v) * 257 + cb] = c[v];
        }
        __syncthreads();

        const float w = (it > ITERS - 10) ? (1.0f / (float)(it - (ITERS - 10))) : 0.0f;
#pragma unroll
        for (int r = 0; r < 16; ++r) {
            float M = m1s[r * 257 + tid];
            float u = u1r[r];
            u += GAMMA * ((b1r[r] - u) - M);
            u1r[r] = u;
            float a = tdq(u);
            cnt1r[r] += (a != 0.0f) ? 1.0f : 0.0f;
            dec1r[r] += (a - dec1r[r]) * w;
            a1s[r * 257 + tid] = a;
        }
        __syncthreads();

        // ---- layer 2: S = (a1/SCALE) @ D2 ; G = a2 @ G2 ----
        if (wave < 4) {
            const int nt = wave;
            const int cb = nt * 16 + n_b;
            v8f c = (v8f)0.0f;
#pragma unroll 4
            for (int k0 = 0; k0 < N1; k0 += 4) {
                v2f a, b;
                a[0] = a1s[m_a * 257 + k0 + kb] * INV_SCALE;
                a[1] = a1s[m_a * 257 + k0 + kb + 1] * INV_SCALE;
                b[0] = d2g[(k0 + kb) * N2 + cb];
                b[1] = d2g[(k0 + kb + 1) * N2 + cb];
                c = wmma4(a, b, c);
            }
#pragma unroll
            for (int v = 0; v < 8; ++v)
                m2s[((lane >> 4) * 8 + v) * 65 + cb] = c[v];
        } else {
            const int nt = wave - 4;
            const int cb = nt * 16 + n_b;
            v8f c = (v8f)0.0f;
#pragma unroll
            for (int k0 = 0; k0 < N2; k0 += 4) {
                v2f a, b;
                a[0] = a2s[m_a * 65 + k0 + kb];
                a[1] = a2s[m_a * 65 + k0 + kb + 1];
                b[0] = g2[(k0 + kb) * N2 + cb];
                b[1] = g2[(k0 + kb + 1) * N2 + cb];
                c = wmma4(a, b, c);
            }
#pragma unroll
            for (int v = 0; v < 8; ++v)
                m2b[((lane >> 4) * 8 + v) * 65 + cb] = c[v];
        }
        __syncthreads();

#pragma unroll
        for (int j = 0; j < 4; ++j) {
            const int r = rq * 4 + j;
            float M = m2s[r * 65 + c2] - m2b[r * 65 + c2];
            float u = u2r[j];
            u += GAMMA * (M - u);
            u2r[j] = u;
            float a = tdq(u);
            cnt2r[j] += (a != 0.0f) ? 1.0f : 0.0f;
            dec2r[j] += (a - dec2r[j]) * w;
            a2s[r * 65 + c2] = a;
        }
        __syncthreads();
    }

    // ---- epilogue ----
#pragma unroll
    for (int r = 0; r < 16; ++r) {
        dec1o[(row0 + r) * N1 + tid] = dec1r[r];
        a1w[(row0 + r) * N1 + tid]   = a1s[r * 257 + tid];
        m1s[r * 257 + tid] = cnt1r[r];
    }
#pragma unroll
    for (int j = 0; j < 4; ++j) {
        const int r = rq * 4 + j;
        dec2o[(row0 + r) * N2 + c2] = dec2r[j];
        m2s[r * 65 + c2] = cnt2r[j];
    }
    for (int idx = tid; idx < 16 * 64; idx += 256)
        a2w[row0 * N2 + idx] = a2s[(idx >> 6) * 65 + (idx & 63)];
    __syncthreads();
    if (tid < 16) {
        float s1 = 0.0f;
        for (int c = 0; c < N1; ++c) s1 += m1s[tid * 257 + c];
        ns1o[row0 + tid] = s1;
        float s2 = 0.0f;
        for (int c = 0; c < N2; ++c) s2 += m2s[tid * 65 + c];
        ns2o[row0 + tid] = s2;
    }
}

// ---------------- epilogue kernels (dictionary update) ----------------

__global__ void res1_kernel(const float* __restrict__ x, const float* __restrict__ a1w,
                            const float* __restrict__ d1t, float* __restrict__ res1) {
    int idx = blockIdx.x * blockDim.x + threadIdx.x;   // 4096*128
    int s = idx >> 7, i = idx & 127;
    float acc = 0.0f;
    for (int j = 0; j < N1; ++j)
        acc += a1w[s * N1 + j] * d1t[j * IN + i];
    res1[idx] = x[idx] - acc;
}

__global__ void res2_kernel(const float* __restrict__ a1w, const float* __restrict__ a2w,
                            const float* __restrict__ d2t, float* __restrict__ res2) {
    int idx = blockIdx.x * blockDim.x + threadIdx.x;   // 4096*256
    int s = idx >> 8, i = idx & 255;
    float acc = 0.0f;
    for (int j = 0; j < N2; ++j)
        acc += a2w[s * N2 + j] * d2t[j * N1 + i];
    res2[idx] = a1w[idx] * INV_SCALE - acc;
}

__global__ __launch_bounds__(128) void dict1_kernel(
    const float* __restrict__ res1, const float* __restrict__ a1w,
    const float* __restrict__ d1, float* __restrict__ d1n) {
    const int j = blockIdx.x;        // 256 columns
    const int i = threadIdx.x;       // 128 rows
    float acc = 0.0f;
    for (int s = 0; s < BATCH; ++s)
        acc += res1[s * IN + i] * a1w[s * N1 + j];
    float val = d1[i * N1 + j] + LR_OVER_B * acc;
    __shared__ float red[128];
    red[i] = val * val;
    __syncthreads();
    for (int off = 64; off > 0; off >>= 1) {
        if (i < off) red[i] += red[i + off];
        __syncthreads();
    }
    float norm = sqrtf(red[0]);
    d1n[i * N1 + j] = val / (norm + 1e-8f);
}

__global__ __launch_bounds__(256) void dict2_kernel(
    const float* __restrict__ res2, const float* __restrict__ a2w,
    const float* __restrict__ d2, float* __restrict__ d2n) {
    const int j = blockIdx.x;        // 64 columns
    const int i = threadIdx.x;       // 256 rows
    float acc = 0.0f;
    for (int s = 0; s < BATCH; ++s)
        acc += res2[s * N1 + i] * a2w[s * N2 + j];
    float val = d2[i * N2 + j] + LR_OVER_B * acc;
    val = fmaxf(val, 0.0f);          // positive dictionary, clamp before normalize
    __shared__ float red[256];
    red[i] = val * val;
    __syncthreads();
    for (int off = 128; off > 0; off >>= 1) {
        if (i < off) red[i] += red[i + off];
        __syncthreads();
    }
    float norm = sqrtf(red[0]);
    d2n[i * N2 + j] = val / (norm + 1e-8f);
}

// ---------------- launch ----------------

extern "C" void kernel_launch(void* const* d_in, const int* in_sizes, int n_in,
                              void* d_out, int out_size, void* d_ws, size_t ws_size,
                              hipStream_t stream) {
    (void)in_sizes; (void)n_in; (void)out_size; (void)ws_size;
    const float* x  = (const float*)d_in[0];   // [4096,128]
    const float* d1 = (const float*)d_in[1];   // [128,256]
    const float* d2 = (const float*)d_in[2];   // [256,64]
    float* out = (float*)d_out;
    float* ws  = (float*)d_ws;

    // workspace layout (floats)
    float* g1   = ws;                 // 65536
    float* g2   = ws + 65536;         // 4096
    float* b1   = ws + 69632;         // 1048576
    float* a1w  = ws + 1118208;       // 1048576
    float* a2w  = ws + 2166784;       // 262144
    float* res1 = ws + 2428928;       // 524288
    float* res2 = ws + 2953216;       // 1048576
    float* d1t  = ws + 4001792;       // 32768
    float* d2t  = ws + 4034560;       // 16384

    // output layout (floats): dec1, dec2, ns1, ns2, D1n, D2n
    float* dec1o = out;
    float* dec2o = out + 1048576;
    float* ns1o  = out + 1310720;
    float* ns2o  = out + 1314816;
    float* d1no  = out + 1318912;
    float* d2no  = out + 1351680;

    gram1_kernel<<<256, 256, 0, stream>>>(d1, g1);
    gram2_kernel<<<16, 256, 0, stream>>>(d2, g2);
    b1_kernel<<<4096, 256, 0, stream>>>(x, d1, b1);
    t1_kernel<<<128, 256, 0, stream>>>(d1, d1t);
    t2_kernel<<<64, 256, 0, stream>>>(d2, d2t);

    nss_main_kernel<<<256, 256, 0, stream>>>(g1, g2, b1, d2,
                                             dec1o, dec2o, ns1o, ns2o, a1w, a2w);

    res1_kernel<<<2048, 256, 0, stream>>>(x, a1w, d1t, res1);
    res2_kernel<<<4096, 256, 0, stream>>>(a1w, a2w, d2t, res2);
    dict1_kernel<<<256, 128, 0, stream>>>(res1, a1w, d1, d1no);
    dict2_kernel<<<64, 256, 0, stream>>>(res2, a2w, d2, d2no);
}